// AMP_GNN_10651518894615
// MI455X (gfx1250) — compile-verified
//
#include <hip/hip_runtime.h>
#include <hip/hip_bf16.h>
#include <math.h>

// ---------------------------------------------------------------------------
// Problem constants (match reference)
// ---------------------------------------------------------------------------
#define BB 16
#define MM 256
#define NN 512
#define UU 8
#define H1C 16
#define H2C 8
#define TT 10
#define LL 2
#define OO 4

typedef __attribute__((ext_vector_type(2))) float v2f;
typedef __attribute__((ext_vector_type(8))) float v8f;

__device__ inline v8f wmma_f32_k4(v2f a, v2f b, v8f c) {
    // D = A(16x4) * B(4x16) + C(16x16), fp32, wave32
    return __builtin_amdgcn_wmma_f32_16x16x4_f32(
        /*neg_a=*/false, a, /*neg_b=*/false, b,
        /*c_mod=*/(short)0, c, /*reuse_a=*/false, /*reuse_b=*/false);
}

__device__ inline void lds_fence() {
    asm volatile("s_wait_dscnt 0" ::: "memory");
}

// ---------------------------------------------------------------------------
// Init: x_hat=0, v_hat=N/M=2, Z=y, V_old=1, g=0       (131072 threads)
// ---------------------------------------------------------------------------
__global__ void init_kernel(const float* __restrict__ y, float* xhat, float* vhat,
                            float* Z, float* Vold, float* g) {
    int i = blockIdx.x * blockDim.x + threadIdx.x;
    if (i < BB * NN) { xhat[i] = 0.0f; vhat[i] = 2.0f; }
    if (i < BB * MM) { Z[i] = y[i]; Vold[i] = 1.0f; }
    g[i] = 0.0f; // BB*NN*H1C = 131072 == grid size
}

// ---------------------------------------------------------------------------
// G[b] = A[b]^T A[b]  via WMMA f32 16x16x4.  One wave per 16x16 output tile.
// grid: 2048 blocks x 256 threads = 16384 waves = B * (N/16)^2 tiles
// ---------------------------------------------------------------------------
__global__ __launch_bounds__(256) void gram_kernel(const float* __restrict__ A,
                                                   float* __restrict__ G) {
    const int lane = threadIdx.x & 31;
    const int w = threadIdx.x >> 5;
    const int lo = lane & 15, hi = lane >> 4;
    const int t = blockIdx.x * 8 + w;
    const int b = t >> 10;            // 1024 tiles per batch
    const int rem = t & 1023;
    const int i0 = (rem >> 5) * 16;
    const int j0 = (rem & 31) * 16;
    const float* Ab = A + (size_t)b * MM * NN;
    v8f c;
#pragma unroll
    for (int r = 0; r < 8; ++r) c[r] = 0.0f;
    for (int k0 = 0; k0 < MM; k0 += 4) {
        const int k = k0 + 2 * hi;
        v2f a, bb;
        a[0]  = Ab[(size_t)k * NN + i0 + lo];
        a[1]  = Ab[(size_t)(k + 1) * NN + i0 + lo];
        bb[0] = Ab[(size_t)k * NN + j0 + lo];
        bb[1] = Ab[(size_t)(k + 1) * NN + j0 + lo];
        c = wmma_f32_k4(a, bb, c);
    }
    float* Gb = G + (size_t)b * NN * NN;
#pragma unroll
    for (int r = 0; r < 8; ++r)
        Gb[(size_t)(i0 + r + 8 * hi) * NN + j0 + lo] = c[r];
}

// ---------------------------------------------------------------------------
// AMP row pass: per (b,m):  V = (A^2)v_hat, Ax = A x_hat, then fused epilogue
// producing Z_new, V_old', inv_denom, scaled_resid.   One wave per row.
// ---------------------------------------------------------------------------
__global__ __launch_bounds__(256) void amp1_kernel(
    const float* __restrict__ A, const float* __restrict__ y,
    const float* __restrict__ s2, const float* __restrict__ xhat,
    const float* __restrict__ vhat, float* Z, float* Vold,
    float* invd, float* scres) {
    const int lane = threadIdx.x & 31;
    const int bm = blockIdx.x * 8 + (threadIdx.x >> 5);
    const int b = bm >> 8;
    const float* Ar = A + (size_t)bm * NN;
    const float* xh = xhat + b * NN;
    const float* vh = vhat + b * NN;
    float av = 0.0f, ax = 0.0f;
    for (int n = lane; n < NN; n += 32) {
        float a = Ar[n];
        av += a * a * vh[n];
        ax += a * xh[n];
    }
#pragma unroll
    for (int off = 16; off > 0; off >>= 1) {
        av += __shfl_xor(av, off, 32);
        ax += __shfl_xor(ax, off, 32);
    }
    if (lane == 0) {
        const float V = av, s = s2[b];
        const float zold = Z[bm], yv = y[bm], vold = Vold[bm];
        const float resid = yv - zold;
        const float znew = ax - (V / (s + vold)) * resid;
        const float id = 1.0f / (s + V);
        Z[bm] = znew;
        Vold[bm] = V;
        invd[bm] = id;
        scres[bm] = (yv - znew) * id;
    }
}

// ---------------------------------------------------------------------------
// AMP column pass: per (b,n): Sigma_inv = (A^2)^T inv_denom, Ar = A^T scaled
// grid (N/32, B), block (32,8); coalesced A reads across tx.
// ---------------------------------------------------------------------------
__global__ void amp2_kernel(const float* __restrict__ A, const float* __restrict__ invd,
                            const float* __restrict__ scres, const float* __restrict__ xhat,
                            float* Sig, float* rvec) {
    __shared__ float ssi[8][32];
    __shared__ float sar[8][32];
    const int tx = threadIdx.x, ty = threadIdx.y;
    const int col = blockIdx.x * 32 + tx;
    const int b = blockIdx.y;
    float si = 0.0f, ar = 0.0f;
    for (int m = ty; m < MM; m += 8) {
        float a = A[((size_t)b * MM + m) * NN + col];
        si += a * a * invd[b * MM + m];
        ar += a * scres[b * MM + m];
    }
    ssi[ty][tx] = si;
    sar[ty][tx] = ar;
    __syncthreads();
    if (ty == 0) {
#pragma unroll
        for (int k = 1; k < 8; ++k) { si += ssi[k][tx]; ar += sar[k][tx]; }
        const float S = 1.0f / (si + 1e-8f);
        Sig[b * NN + col] = S;
        rvec[b * NN + col] = xhat[b * NN + col] + S * ar;
    }
}

// ---------------------------------------------------------------------------
// t==0 only: u = init_lin([A^T y, diag(G), sigma2])
// ---------------------------------------------------------------------------
__global__ void initu_kernel(const float* __restrict__ A, const float* __restrict__ y,
                             const float* __restrict__ G, const float* __restrict__ s2,
                             const float* __restrict__ iw, const float* __restrict__ ib,
                             float* u) {
    __shared__ float red[8][32];
    const int tx = threadIdx.x, ty = threadIdx.y;
    const int col = blockIdx.x * 32 + tx;
    const int b = blockIdx.y;
    float aty = 0.0f;
    for (int m = ty; m < MM; m += 8)
        aty += A[((size_t)b * MM + m) * NN + col] * y[b * MM + m];
    red[ty][tx] = aty;
    __syncthreads();
    if (ty == 0) {
#pragma unroll
        for (int k = 1; k < 8; ++k) aty += red[k][tx];
        const float dg = G[(size_t)b * NN * NN + (size_t)col * NN + col];
        const float s = s2[b];
#pragma unroll
        for (int f = 0; f < UU; ++f)
            u[((size_t)b * NN + col) * UU + f] =
                ib[f] + iw[f * 3 + 0] * aty + iw[f * 3 + 1] * dg + iw[f * 3 + 2] * s;
    }
}

// ---------------------------------------------------------------------------
// HOT KERNEL: pairwise edge MLP + masked sum over j.  One wave per (b,n) row.
// Layer1 folds u_n / sigma2 into a per-row effective bias (K: 18 -> 9).
// 3+4+2 = 9 WMMA f32 16x16x4 per 16-j tile, 32 tiles per row.
// LDS used for D-layout -> A-layout relayout between layers (wave-private).
// ---------------------------------------------------------------------------
__global__ __launch_bounds__(256) void edge_mlp_kernel(
    const float* __restrict__ u, const float* __restrict__ G,
    const float* __restrict__ sigma2,
    const float* __restrict__ w1, const float* __restrict__ b1,
    const float* __restrict__ w2, const float* __restrict__ b2,
    const float* __restrict__ w3, const float* __restrict__ b3,
    float* __restrict__ m_sum) {
    __shared__ float lds[8][256]; // 16x16 scratch tile per wave
    const int lane = threadIdx.x & 31;
    const int w = threadIdx.x >> 5;
    const int lo = lane & 15, hi = lane >> 4;
    const int row = blockIdx.x * 8 + w;
    const int b = row >> 9;
    const int n = row & (NN - 1);
    const float s2 = sigma2[b];
    float* L = lds[w];

    // per-row effective layer-1 bias for hidden unit `lo`
    float be = b1[lo];
    {
        const float* un = u + (size_t)row * UU;
        const float* w1r = w1 + lo * 18;
#pragma unroll
        for (int f = 0; f < UU; ++f) be += w1r[f] * un[f];
        be += w1r[17] * s2;
    }
    // layer-1 B operand chunks (edge features: 0..7 = u_j -> pin[8+k], 8 = G -> pin[16])
    v2f B1[3];
#pragma unroll
    for (int c = 0; c < 3; ++c)
#pragma unroll
        for (int v = 0; v < 2; ++v) {
            const int k = 4 * c + 2 * hi + v;
            float val = 0.0f;
            if (k < 8) val = w1[lo * 18 + 8 + k];
            else if (k == 8) val = w1[lo * 18 + 16];
            B1[c][v] = val;
        }
    // layer-2 B operand chunks (W2: [8][16], zero-pad cols 8..15)
    v2f B2[4];
#pragma unroll
    for (int c = 0; c < 4; ++c)
#pragma unroll
        for (int v = 0; v < 2; ++v) {
            const int k = 4 * c + 2 * hi + v;
            B2[c][v] = (lo < H2C) ? w2[lo * H1C + k] : 0.0f;
        }
    // layer-3 B operand chunks (W3: [8][8])
    v2f B3[2];
#pragma unroll
    for (int c = 0; c < 2; ++c)
#pragma unroll
        for (int v = 0; v < 2; ++v) {
            const int k = 4 * c + 2 * hi + v;
            B3[c][v] = (lo < UU) ? w3[lo * H2C + k] : 0.0f;
        }
    const float b2v = (lo < H2C) ? b2[lo] : 0.0f;

    float macc = 0.0f;
    const float* Grow = G + (size_t)b * NN * NN + (size_t)n * NN;

    for (int jt = 0; jt < NN / 16; ++jt) {
        const int j0 = jt * 16;
        const float* uj = u + ((size_t)b * NN + j0 + lo) * UU;
        const float gv = Grow[j0 + lo];

        // ---- layer 1: h1 = relu(pin_edge @ W1_edge^T + bias_eff) ----
        v8f c1;
#pragma unroll
        for (int r = 0; r < 8; ++r) c1[r] = be;
        v2f a;
        a[0] = uj[2 * hi];     a[1] = uj[2 * hi + 1];
        c1 = wmma_f32_k4(a, B1[0], c1);
        a[0] = uj[4 + 2 * hi]; a[1] = uj[5 + 2 * hi];
        c1 = wmma_f32_k4(a, B1[1], c1);
        a[0] = (hi == 0) ? gv : 0.0f; a[1] = 0.0f;
        c1 = wmma_f32_k4(a, B1[2], c1);
#pragma unroll
        for (int r = 0; r < 8; ++r) c1[r] = fmaxf(c1[r], 0.0f);

        // relayout h1: D layout -> LDS -> A layout
#pragma unroll
        for (int r = 0; r < 8; ++r) L[(r + 8 * hi) * 16 + lo] = c1[r];
        lds_fence();

        // ---- layer 2 ----
        v8f c2;
#pragma unroll
        for (int r = 0; r < 8; ++r) c2[r] = b2v;
#pragma unroll
        for (int c = 0; c < 4; ++c) {
            v2f aa;
            aa[0] = L[lo * 16 + 4 * c + 2 * hi];
            aa[1] = L[lo * 16 + 4 * c + 2 * hi + 1];
            c2 = wmma_f32_k4(aa, B2[c], c2);
        }
#pragma unroll
        for (int r = 0; r < 8; ++r) c2[r] = fmaxf(c2[r], 0.0f);

#pragma unroll
        for (int r = 0; r < 8; ++r) L[(r + 8 * hi) * 16 + lo] = c2[r];
        lds_fence();

        // ---- layer 3 (no bias here; +511*b3 added once at the end) ----
        v8f c3;
#pragma unroll
        for (int r = 0; r < 8; ++r) c3[r] = 0.0f;
#pragma unroll
        for (int c = 0; c < 2; ++c) {
            v2f aa;
            aa[0] = L[lo * 16 + 4 * c + 2 * hi];
            aa[1] = L[lo * 16 + 4 * c + 2 * hi + 1];
            c3 = wmma_f32_k4(aa, B3[c], c3);
        }
        // masked reduction over the 16 j's of this tile (skip j == n)
#pragma unroll
        for (int r = 0; r < 8; ++r) {
            const int jj = j0 + r + 8 * hi;
            macc += (jj == n) ? 0.0f : c3[r];
        }
    }
    macc += __shfl_xor(macc, 16, 32);
    if (lane < UU)
        m_sum[(size_t)row * UU + lane] = macc + (float)(NN - 1) * b3[lane];
}

// ---------------------------------------------------------------------------
// GRU cell, gate order (r,z,n).  One thread per (row, hidden unit).
// ---------------------------------------------------------------------------
__global__ void gru_kernel(const float* __restrict__ msum, const float* __restrict__ rv,
                           const float* __restrict__ Sg, const float* __restrict__ gin,
                           float* __restrict__ gout,
                           const float* __restrict__ wih, const float* __restrict__ whh,
                           const float* __restrict__ bih, const float* __restrict__ bhh) {
    const int idx = blockIdx.x * blockDim.x + threadIdx.x; // B*N*H1
    const int row = idx >> 4, un = idx & 15;
    float x[UU + 2];
#pragma unroll
    for (int f = 0; f < UU; ++f) x[f] = msum[(size_t)row * UU + f];
    x[UU] = rv[row];
    x[UU + 1] = Sg[row];
    float h[H1C];
#pragma unroll
    for (int k = 0; k < H1C; ++k) h[k] = gin[(size_t)row * H1C + k];
    float gi[3], gh[3];
#pragma unroll
    for (int g3 = 0; g3 < 3; ++g3) {
        const int r0 = g3 * H1C + un;
        float a = bih[r0];
#pragma unroll
        for (int f = 0; f < UU + 2; ++f) a += wih[r0 * (UU + 2) + f] * x[f];
        gi[g3] = a;
        float c = bhh[r0];
#pragma unroll
        for (int k = 0; k < H1C; ++k) c += whh[r0 * H1C + k] * h[k];
        gh[g3] = c;
    }
    const float rg = 1.0f / (1.0f + expf(-(gi[0] + gh[0])));
    const float zg = 1.0f / (1.0f + expf(-(gi[1] + gh[1])));
    const float ng = tanhf(gi[2] + rg * gh[2]);
    gout[idx] = (1.0f - zg) * ng + zg * h[un];
}

// ---------------------------------------------------------------------------
// u = upd_w @ g + upd_b.  One thread per (row, out unit).
// ---------------------------------------------------------------------------
__global__ void upd_kernel(const float* __restrict__ g, const float* __restrict__ w,
                           const float* __restrict__ bb, float* __restrict__ u) {
    const int idx = blockIdx.x * blockDim.x + threadIdx.x; // B*N*U
    const int row = idx >> 3, o = idx & 7;
    float a = bb[o];
#pragma unroll
    for (int k = 0; k < H1C; ++k) a += w[o * H1C + k] * g[(size_t)row * H1C + k];
    u[idx] = a;
}

// ---------------------------------------------------------------------------
// Readout MLP + softmax + moments.  One thread per (b,n).
// ---------------------------------------------------------------------------
__global__ void readout_kernel(const float* __restrict__ u,
                               const float* __restrict__ w1, const float* __restrict__ b1,
                               const float* __restrict__ w2, const float* __restrict__ b2,
                               const float* __restrict__ w3, const float* __restrict__ b3,
                               float* __restrict__ xhat, float* __restrict__ vhat) {
    const int i = blockIdx.x * blockDim.x + threadIdx.x; // B*N
    float uu[UU];
#pragma unroll
    for (int f = 0; f < UU; ++f) uu[f] = u[(size_t)i * UU + f];
    float h1[H1C];
#pragma unroll
    for (int h = 0; h < H1C; ++h) {
        float a = b1[h];
#pragma unroll
        for (int f = 0; f < UU; ++f) a += w1[h * UU + f] * uu[f];
        h1[h] = fmaxf(a, 0.0f);
    }
    float h2[H2C];
#pragma unroll
    for (int o = 0; o < H2C; ++o) {
        float a = b2[o];
#pragma unroll
        for (int k = 0; k < H1C; ++k) a += w2[o * H1C + k] * h1[k];
        h2[o] = fmaxf(a, 0.0f);
    }
    float lg[OO];
#pragma unroll
    for (int o = 0; o < OO; ++o) {
        float a = b3[o];
#pragma unroll
        for (int k = 0; k < H2C; ++k) a += w3[o * H2C + k] * h2[k];
        lg[o] = a;
    }
    const float mx = fmaxf(fmaxf(lg[0], lg[1]), fmaxf(lg[2], lg[3]));
    float e[OO], s = 0.0f;
#pragma unroll
    for (int o = 0; o < OO; ++o) { e[o] = expf(lg[o] - mx); s += e[o]; }
    const float cst[OO] = {-3.0f, -1.0f, 1.0f, 3.0f};
    float xh = 0.0f, x2 = 0.0f;
#pragma unroll
    for (int o = 0; o < OO; ++o) {
        const float p = e[o] / s;
        xh += p * cst[o];
        x2 += p * cst[o] * cst[o];
    }
    xhat[i] = xh;
    vhat[i] = x2 - xh * xh;
}

// ---------------------------------------------------------------------------
// Host orchestration
// ---------------------------------------------------------------------------
extern "C" void kernel_launch(void* const* d_in, const int* in_sizes, int n_in,
                              void* d_out, int out_size, void* d_ws, size_t ws_size,
                              hipStream_t stream) {
    const float* y      = (const float*)d_in[0];
    const float* A      = (const float*)d_in[1];
    const float* s2     = (const float*)d_in[2];
    const float* init_w = (const float*)d_in[3];
    const float* init_b = (const float*)d_in[4];
    const float* p1_w   = (const float*)d_in[5];
    const float* p1_b   = (const float*)d_in[6];
    const float* p2_w   = (const float*)d_in[7];
    const float* p2_b   = (const float*)d_in[8];
    const float* p3_w   = (const float*)d_in[9];
    const float* p3_b   = (const float*)d_in[10];
    const float* gwih   = (const float*)d_in[11];
    const float* gwhh   = (const float*)d_in[12];
    const float* gbih   = (const float*)d_in[13];
    const float* gbhh   = (const float*)d_in[14];
    const float* updw   = (const float*)d_in[15];
    const float* updb   = (const float*)d_in[16];
    const float* r1w    = (const float*)d_in[17];
    const float* r1b    = (const float*)d_in[18];
    const float* r2w    = (const float*)d_in[19];
    const float* r2b    = (const float*)d_in[20];
    const float* r3w    = (const float*)d_in[21];
    const float* r3b    = (const float*)d_in[22];

    float* ws = (float*)d_ws;
    float* G     = ws; ws += (size_t)BB * NN * NN;   // 16.8 MB
    float* u     = ws; ws += (size_t)BB * NN * UU;
    float* g0    = ws; ws += (size_t)BB * NN * H1C;
    float* g1    = ws; ws += (size_t)BB * NN * H1C;
    float* msum  = ws; ws += (size_t)BB * NN * UU;
    float* Vold  = ws; ws += (size_t)BB * MM;
    float* Zb    = ws; ws += (size_t)BB * MM;
    float* invd  = ws; ws += (size_t)BB * MM;
    float* scres = ws; ws += (size_t)BB * MM;
    float* Sig   = ws; ws += (size_t)BB * NN;
    float* rvec  = ws; ws += (size_t)BB * NN;
    float* vhat  = ws; ws += (size_t)BB * NN;
    float* xhat  = (float*)d_out; // final output == last x_hat

    init_kernel<<<512, 256, 0, stream>>>(y, xhat, vhat, Zb, Vold, g0);
    gram_kernel<<<2048, 256, 0, stream>>>(A, G);

    float* gin = g0;
    float* gout = g1;
    for (int t = 0; t < TT; ++t) {
        amp1_kernel<<<512, 256, 0, stream>>>(A, y, s2, xhat, vhat, Zb, Vold, invd, scres);
        amp2_kernel<<<dim3(NN / 32, BB), dim3(32, 8), 0, stream>>>(A, invd, scres, xhat, Sig, rvec);
        if (t == 0)
            initu_kernel<<<dim3(NN / 32, BB), dim3(32, 8), 0, stream>>>(A, y, G, s2, init_w, init_b, u);
        for (int l = 0; l < LL; ++l) {
            edge_mlp_kernel<<<1024, 256, 0, stream>>>(u, G, s2, p1_w, p1_b, p2_w, p2_b,
                                                      p3_w, p3_b, msum);
            gru_kernel<<<512, 256, 0, stream>>>(msum, rvec, Sig, gin, gout,
                                                gwih, gwhh, gbih, gbhh);
            upd_kernel<<<256, 256, 0, stream>>>(gout, updw, updb, u);
            float* tmp = gin; gin = gout; gout = tmp;
        }
        readout_kernel<<<BB * NN / 256, 256, 0, stream>>>(u, r1w, r1b, r2w, r2b, r3w, r3b,
                                                          xhat, vhat);
    }
}